// ButterflyRotationLayer_4827543241363
// MI455X (gfx1250) — compile-verified
//
#include <hip/hip_runtime.h>
#include <hip/hip_bf16.h>

// Closed-form butterfly product for MI455X (gfx1250, wave32).
// R[i][j] = prod over 12 stages of {cos, +/-sin}(theta[m][t]) along the unique
// butterfly path i -> j. Factorized per 64x64 tile as diag(a) @ B64 so the
// final combine runs on v_wmma_f32_16x16x4_f32 while the kernel stays
// HBM-store-bound (64 MB out @ 23.3 TB/s ~ 3 us).

#define D        4096
#define HALFD    2048
#define NSTAGE   12
#define NB       64            // number of 64-wide blocks per dim
#define ATAB_ELT (64 * 64 * 64)   // a[iH][jH][iL]
#define BTAB_ELT (64 * 64 * 64)   // B64[jH][iL][jL]

typedef __attribute__((ext_vector_type(2))) float v2f;
typedef __attribute__((ext_vector_type(8))) float v8f;

// Per-element path factor for stage with bit b.
__device__ __forceinline__ float stage_factor(const float* __restrict__ thetas,
                                              int m, int b, int i, int j) {
    int t = ((j >> (b + 1)) << b) | (i & ((1 << b) - 1));
    float th = thetas[m * HALFD + t];
    float s, c;
    __sincosf(th, &s, &c);
    int ib = (i >> b) & 1;
    int jb = (j >> b) & 1;
    // (ib==jb) -> c ; (1,0) -> s ; (0,1) -> -s
    return (ib == jb) ? c : (ib ? s : -s);
}

// Phase 1a: row scales a[iH][jH][iL] = product of stages m=0..5 (bits 11..6).
// For b >= 6 neither t nor the sign depends on jL, so j = jH*64 suffices.
__global__ void bfly_scales(const float* __restrict__ thetas,
                            float* __restrict__ aTab) {
    int idx = blockIdx.x * blockDim.x + threadIdx.x;   // 0 .. 262143
    int iL = idx & 63;
    int jH = (idx >> 6) & 63;
    int iH = idx >> 12;
    int i = iH * 64 + iL;
    int j = jH * 64;
    float prod = 1.0f;
#pragma unroll
    for (int m = 0; m < 6; ++m) {
        prod *= stage_factor(thetas, m, 11 - m, i, j);
    }
    aTab[idx] = prod;   // layout [iH][jH][iL]
}

// Phase 1b: dense blocks B64[jH][iL][jL] = product of stages m=6..11 (bits 5..0).
// For b <= 5 neither t nor the sign depends on iH, so i = iL suffices.
__global__ void bfly_blocks(const float* __restrict__ thetas,
                            float* __restrict__ bTab) {
    int idx = blockIdx.x * blockDim.x + threadIdx.x;   // 0 .. 262143
    int jL = idx & 63;
    int iL = (idx >> 6) & 63;
    int jH = idx >> 12;
    int j = jH * 64 + jL;
    float prod = 1.0f;
#pragma unroll
    for (int m = 6; m < NSTAGE; ++m) {
        prod *= stage_factor(thetas, m, 11 - m, iL, j);
    }
    bTab[idx] = prod;   // layout [jH][iL][jL]
}

// Phase 2: one workgroup (256 threads = 8 wave32) per 64x64 output tile.
// R_tile = diag(a) @ B64[jH], computed as chained v_wmma_f32_16x16x4_f32.
__global__ void __launch_bounds__(256)
bfly_tiles(const float* __restrict__ aTab,
           const float* __restrict__ bTab,
           float* __restrict__ out) {
    __shared__ float lB[64 * 64];   // 16 KB of 320 KB/WGP
    __shared__ float lA[64];

    const int jH = blockIdx.x & 63;
    const int iH = blockIdx.x >> 6;

    // Stage B64[jH] (16 KB) into LDS: 256 lanes x 4 x b128.
    {
        const float4* __restrict__ src =
            (const float4*)(bTab + (size_t)jH * 4096);
        float4* dst = (float4*)lB;
#pragma unroll
        for (int r = 0; r < 4; ++r) {
            int o = threadIdx.x + 256 * r;
            dst[o] = src[o];
        }
        if (threadIdx.x < 64) {
            lA[threadIdx.x] = aTab[((size_t)iH * 64 + jH) * 64 + threadIdx.x];
        }
    }
    __syncthreads();

    const int lane = threadIdx.x & 31;
    const int wv   = threadIdx.x >> 5;   // 8 waves, 2 subtiles each
    const int nlo  = lane & 15;          // M (A rows / C cols) index
    const int hi   = lane >> 4;          // lane half select

#pragma unroll
    for (int s = 0; s < 2; ++s) {
        const int st = wv * 2 + s;       // 16 subtiles of 16x16
        const int iT = st >> 2;
        const int jT = st & 3;

        const float am = lA[iT * 16 + nlo];   // diag entry for row m=nlo
        v8f c = {0.f, 0.f, 0.f, 0.f, 0.f, 0.f, 0.f, 0.f};

#pragma unroll
        for (int kc = 0; kc < 4; ++kc) {
            // A fragment (16x4 f32): lanes 0-15 hold K=kc*4+{0,1},
            // lanes 16-31 hold K=kc*4+{2,3} (ISA 7.12.2). Diagonal block:
            // A[m,k] = (m==k) ? a[m] : 0, block-local k = kc*4 + kk.
            const int k0 = kc * 4 + hi * 2;
            v2f va, vb;
            va.x = (nlo == k0)     ? am : 0.0f;
            va.y = (nlo == k0 + 1) ? am : 0.0f;
            // B fragment (4x16 f32): rows striped across lanes within a VGPR,
            // mirroring the A half-split: v.x row k0, v.y row k0+1, col = nlo.
            vb.x = lB[(iT * 16 + k0) * 64 + jT * 16 + nlo];
            vb.y = lB[(iT * 16 + k0 + 1) * 64 + jT * 16 + nlo];
            c = __builtin_amdgcn_wmma_f32_16x16x4_f32(
                    false, va, false, vb, (short)0, c, false, false);
        }

        // C/D layout: VGPR r -> M=r (lanes 0-15) / M=r+8 (lanes 16-31), N=nlo.
        const int col = jH * 64 + jT * 16 + nlo;
        const size_t rowBase = (size_t)(iH * 64 + iT * 16 + hi * 8);
#pragma unroll
        for (int r = 0; r < 8; ++r) {
            out[(rowBase + r) * D + col] = c[r];
        }
    }
}

extern "C" void kernel_launch(void* const* d_in, const int* in_sizes, int n_in,
                              void* d_out, int out_size, void* d_ws, size_t ws_size,
                              hipStream_t stream) {
    const float* thetas = (const float*)d_in[0];   // [12, 2048] fp32
    float* out  = (float*)d_out;                   // [4096, 4096] fp32
    float* aTab = (float*)d_ws;                    // 1 MB
    float* bTab = aTab + ATAB_ELT;                 // 1 MB

    bfly_scales<<<ATAB_ELT / 256, 256, 0, stream>>>(thetas, aTab);
    bfly_blocks<<<BTAB_ELT / 256, 256, 0, stream>>>(thetas, bTab);
    bfly_tiles <<<NB * NB,        256, 0, stream>>>(aTab, bTab, out);
}